// AttentionLayer_62809601736967
// MI455X (gfx1250) — compile-verified
//
#include <hip/hip_runtime.h>
#include <math.h>

typedef float v2f __attribute__((ext_vector_type(2)));
typedef float v8f __attribute__((ext_vector_type(8)));

#define NH    16
#define NX    8
#define LSEQ  1024
#define NTHR  256
#define NWAVE 8
#define LPW   (LSEQ / NWAVE)   // 128 sequence positions per wave in phase 1

__global__ __launch_bounds__(NTHR)
void chanattn_kernel(const float* __restrict__ x,
                     const float* __restrict__ pos,
                     const float* __restrict__ Wq,
                     const float* __restrict__ Wk,
                     const float* __restrict__ Wv,
                     const float* __restrict__ Wo,
                     float* __restrict__ out)
{
    __shared__ __align__(16) float s_x[LSEQ * NX];   // 32 KB  x[b] staged
    __shared__ __align__(16) float s_pos[LSEQ];      //  4 KB  pos[b] staged
    __shared__ float s_part[NWAVE * 256];            //  8 KB  per-wave score partials
    __shared__ float s_attn[NH * NH];                // score -> attn (in place)
    __shared__ float s_M[NH * NH];                   // W_o @ attn
    __shared__ float s_G[NH * NX];                   // M @ W_v  (16x8)

    const int b    = blockIdx.x;
    const int tid  = threadIdx.x;
    const int lane = tid & 31;
    const int wave = tid >> 5;
    const int d    = lane & 15;          // head-dim channel this lane owns
    const int koff = (lane >> 4) << 1;   // K-slot base (0 or 2) in WMMA A/B layout

    // ---------------- stage x[b], pos[b] into LDS -----------------------
    {
        const float4* xg = (const float4*)(x + (size_t)b * LSEQ * NX);
        float4* xs = (float4*)s_x;
        #pragma unroll
        for (int i = 0; i < (LSEQ * NX / 4) / NTHR; ++i)   // 8 iters
            xs[tid + i * NTHR] = xg[tid + i * NTHR];
        const float4* pg = (const float4*)(pos + (size_t)b * LSEQ);
        ((float4*)s_pos)[tid] = pg[tid];
    }

    // per-lane rows of W_q / W_k  (row d, 8 floats each; hot in cache)
    float wq[NX], wk[NX];
    #pragma unroll
    for (int j = 0; j < NX; ++j) { wq[j] = Wq[d * NX + j]; wk[j] = Wk[d * NX + j]; }

    __syncthreads();

    // ---------------- phase 1: score = q^T k  via WMMA f32 16x16x4 ------
    // A[m=d, k=l'] = q[l0+k, d],  B[k=l', n=e] = k[l0+k, e]
    // lane layout: m/n = lane&15 ; k = 2*(lane>>4) + v
    const float fm = (float)((d >> 1) + 1) * 3.14159265358979323846f;
    v8f acc = {};
    for (int s = 0; s < LPW / 4; ++s) {                // 32 WMMA per wave
        const int l0 = wave * LPW + s * 4;
        v2f av, bv;
        #pragma unroll
        for (int v = 0; v < 2; ++v) {
            const int l = l0 + koff + v;
            const float* xr = &s_x[l * NX];
            float sn, cs;
            __sincosf(s_pos[l] * fm, &sn, &cs);
            const float P = (d & 1) ? cs : sn;
            float qv = P, kv = P;
            #pragma unroll
            for (int j = 0; j < NX; ++j) {
                qv = fmaf(xr[j], wq[j], qv);
                kv = fmaf(xr[j], wk[j], kv);
            }
            av[v] = qv; bv[v] = kv;
        }
        acc = __builtin_amdgcn_wmma_f32_16x16x4_f32(
                  false, av, false, bv, (short)0, acc, false, false);
    }
    // spill per-wave partial (D layout: lane n=lane&15, VGPR r -> m=r+8*(lane>>4))
    #pragma unroll
    for (int r = 0; r < 8; ++r)
        s_part[wave * 256 + lane * 8 + r] = acc[r];
    __syncthreads();

    // ---------------- reduce partials, scale by 1/sqrt(L)=1/32 ----------
    {
        float sum = 0.f;
        #pragma unroll
        for (int w = 0; w < NWAVE; ++w) sum += s_part[w * 256 + tid];
        const int ln = tid >> 3, r = tid & 7;
        const int m  = r + ((ln >> 4) << 3);
        const int e  = ln & 15;
        s_attn[m * NH + e] = sum * (1.0f / 32.0f);
    }
    __syncthreads();

    // ---------------- softmax over e (rows), 16 threads -----------------
    if (tid < NH) {
        float mx = -3.4e38f;
        #pragma unroll
        for (int e = 0; e < NH; ++e) mx = fmaxf(mx, s_attn[tid * NH + e]);
        float ex[NH]; float sum = 0.f;
        #pragma unroll
        for (int e = 0; e < NH; ++e) {
            ex[e] = __expf(s_attn[tid * NH + e] - mx);
            sum += ex[e];
        }
        const float inv = 1.0f / sum;
        #pragma unroll
        for (int e = 0; e < NH; ++e) s_attn[tid * NH + e] = ex[e] * inv;
    }
    __syncthreads();

    // ---------------- M = W_o @ attn  (256 threads, 1 elem each) --------
    {
        const int h = tid >> 4, e = tid & 15;
        float a = 0.f;
        #pragma unroll
        for (int dd = 0; dd < NH; ++dd)
            a = fmaf(Wo[h * NH + dd], s_attn[dd * NH + e], a);
        s_M[h * NH + e] = a;
    }
    __syncthreads();

    // ---------------- G = M @ W_v  (16x8, 128 threads) ------------------
    if (tid < NH * NX) {
        const int h = tid >> 3, j = tid & 7;
        float a = 0.f;
        #pragma unroll
        for (int e = 0; e < NH; ++e)
            a = fmaf(s_M[h * NH + e], Wv[e * NX + j], a);
        s_G[h * NX + j] = a;
    }
    __syncthreads();

    // ---------------- phase 3: out_tile = x_tile @ G^T via WMMA ---------
    // A[m=l-lbase, k=j] = x[l, j] ; B[k=j, n=h] = G[h, j] ; K=8 as two K=4
    v2f bg0, bg1;
    bg0[0] = s_G[(lane & 15) * NX + koff];
    bg0[1] = s_G[(lane & 15) * NX + koff + 1];
    bg1[0] = s_G[(lane & 15) * NX + koff + 4];
    bg1[1] = s_G[(lane & 15) * NX + koff + 5];

    float* outb = out + (size_t)b * LSEQ * NH;
    for (int t = wave; t < LSEQ / NH; t += NWAVE) {    // 8 tiles per wave
        const int lbase = t * NH;
        const int l = lbase + (lane & 15);
        v2f a0, a1;
        a0[0] = s_x[l * NX + koff];     a0[1] = s_x[l * NX + koff + 1];
        a1[0] = s_x[l * NX + koff + 4]; a1[1] = s_x[l * NX + koff + 5];
        v8f c = {};
        c = __builtin_amdgcn_wmma_f32_16x16x4_f32(
                false, a0, false, bg0, (short)0, c, false, false);
        c = __builtin_amdgcn_wmma_f32_16x16x4_f32(
                false, a1, false, bg1, (short)0, c, false, false);
        // D layout: lane n=lane&15 -> h ; VGPR r -> row lbase + r + 8*(lane>>4)
        #pragma unroll
        for (int r = 0; r < 8; ++r)
            outb[(size_t)(lbase + r + ((lane >> 4) << 3)) * NH + (lane & 15)] = c[r];
    }
}

extern "C" void kernel_launch(void* const* d_in, const int* in_sizes, int n_in,
                              void* d_out, int out_size, void* d_ws, size_t ws_size,
                              hipStream_t stream) {
    (void)n_in; (void)out_size; (void)d_ws; (void)ws_size;
    const float* x   = (const float*)d_in[0];
    const float* pos = (const float*)d_in[1];
    const float* Wq  = (const float*)d_in[2];
    const float* Wk  = (const float*)d_in[3];
    const float* Wv  = (const float*)d_in[4];
    const float* Wo  = (const float*)d_in[5];
    float* outp      = (float*)d_out;
    const int B = in_sizes[1] / LSEQ;   // pos is [B, L]
    chanattn_kernel<<<B, NTHR, 0, stream>>>(x, pos, Wq, Wk, Wv, Wo, outp);
}